// KANLayer_14714557956114
// MI455X (gfx1250) — compile-verified
//
#include <hip/hip_runtime.h>
#include <stdint.h>

// Problem constants (match the reference)
#define BATCH   512
#define INF     256           // in features
#define OUTF    256           // out features
#define NINT    15            // GRID-1 cubic intervals
#define COEF_FLOATS (INF * NINT * 4)   // 15360 f32 = 61440 B per output row
#define BT      256           // batch rows per block (half of BATCH)
#define THREADS 256           // 8 waves (wave32)

typedef unsigned int u32;
typedef u32 __attribute__((ext_vector_type(4))) u32x4;
typedef int  __attribute__((ext_vector_type(4))) i32x4;
typedef int  __attribute__((ext_vector_type(8))) i32x8;

__global__ __launch_bounds__(THREADS)
void kan_layer_kernel(const float* __restrict__ x,      // (B, IN)
                      const float* __restrict__ w,      // (OUT, IN)
                      const float* __restrict__ bias,   // (OUT,)
                      const float* __restrict__ coeffs, // (OUT, IN, 15, 4)
                      float* __restrict__ out)          // (B, OUT)
{
    __shared__ float sC[COEF_FLOATS];   // 60 KB coeff tile for this output feature
    __shared__ float sW[INF];           // this output feature's weight row

    const int o     = (int)(blockIdx.x >> 1);        // output feature
    const int bbase = (int)(blockIdx.x & 1) * BT;    // batch chunk start
    const int tid   = (int)threadIdx.x;
    const int lane  = tid & 31;
    const int wave  = tid >> 5;

    // ---- stage weight row (1 KB), coalesced ----
    sW[tid] = w[(size_t)o * INF + tid];              // THREADS == INF

    // ---- stage 60 KB coeff tile with the Tensor Data Mover (one DMA per block) ----
    if (wave == 0) {
        const uint64_t ga  = (uint64_t)(uintptr_t)(coeffs + (size_t)o * COEF_FLOATS);
        const u32      lds = (u32)(uintptr_t)(&sC[0]);

        // D# group 0: count=1 (valid), lds_addr, global_addr[56:0], type=2 ("image")
        u32x4 g0 = {0u, 0u, 0u, 0u};
        g0.x = 1u;                                        // count = 1, user mode
        g0.y = lds;                                       // LDS byte address
        g0.z = (u32)ga;                                   // global_addr[31:0]
        g0.w = (u32)((ga >> 32) & 0x01FFFFFFu) | (2u << 30); // addr[56:32] | type=2

        // D# group 1: 1-row 2D tensor, data_size=4B, tile = whole 15360-elem row
        i32x8 g1 = {0, 0, 0, 0, 0, 0, 0, 0};
        g1[0] = (int)(2u << 16);                          // data_size = 2 -> 4 bytes
        g1[1] = (int)((COEF_FLOATS & 0xFFFF) << 16);      // tensor_dim0[15:0]  @ bits[63:48]
        g1[2] = (int)((COEF_FLOATS >> 16) | (1u << 16));  // tensor_dim0[31:16] | tensor_dim1=1
        g1[3] = (int)((COEF_FLOATS & 0xFFFF) << 16);      // tile_dim0 = 15360  @ bits[127:112]
        g1[4] = 1;                                        // tile_dim1 = 1
        g1[5] = COEF_FLOATS;                              // tensor_dim0_stride[31:0]
        // g1[6], g1[7] = 0 (stride hi, dim1 stride unused)

        i32x4 g2 = {0, 0, 0, 0};                          // <=2D tensor: groups 2/3 unused
        i32x4 g3 = {0, 0, 0, 0};

#if defined(__clang_major__) && (__clang_major__ >= 23)
        i32x8 g4 = {0, 0, 0, 0, 0, 0, 0, 0};
        __builtin_amdgcn_tensor_load_to_lds(g0, g1, g2, g3, g4, 0);
#else
        __builtin_amdgcn_tensor_load_to_lds(g0, g1, g2, g3, 0);
#endif
        __builtin_amdgcn_s_wait_tensorcnt(0);             // TDM done -> LDS data visible
    }
    __syncthreads();

    const float bo   = bias[o];
    const float invh = 7.5f;          // (GRID-1)/2
    const float h    = 2.0f / 15.0f;  // breakpoint spacing

    // Per-lane weight registers: lane handles i = lane + 32k
    float wreg[8];
#pragma unroll
    for (int k = 0; k < 8; ++k) wreg[k] = sW[lane + 32 * k];

    // Each wave strides over batch rows of this chunk
    for (int r = wave; r < BT; r += THREADS / 32) {
        const int b = bbase + r;
        const float* __restrict__ xrow = x + (size_t)b * INF;
        float acc = 0.0f;
#pragma unroll
        for (int k = 0; k < 8; ++k) {
            const int i  = lane + 32 * k;
            const float wx = wreg[k] * xrow[i];                    // coalesced x load
            const float xc = fminf(1.0f, fmaxf(-1.0f, wx));
            int idx = (int)((xc + 1.0f) * invh);                   // floor, xc>=-1 so >=0
            idx = idx > (NINT - 1) ? (NINT - 1) : idx;             // clamp (xc==1 case)
            const float dx = xc - (-1.0f + h * (float)idx);
            // 16B-aligned LDS gather of the interval's 4 coeffs (ds_load_b128)
            const float4 c = *reinterpret_cast<const float4*>(&sC[(i * NINT + idx) * 4]);
            acc += ((c.w * dx + c.z) * dx + c.y) * dx + c.x;       // Horner
        }
        // wave32 tree reduction over the 32 i-lanes
#pragma unroll
        for (int off = 16; off > 0; off >>= 1)
            acc += __shfl_xor(acc, off, 32);
        if (lane == 0)
            out[(size_t)b * OUTF + o] = acc + bo;
    }
}

extern "C" void kernel_launch(void* const* d_in, const int* in_sizes, int n_in,
                              void* d_out, int out_size, void* d_ws, size_t ws_size,
                              hipStream_t stream) {
    (void)in_sizes; (void)n_in; (void)d_ws; (void)ws_size; (void)out_size;
    const float* x      = (const float*)d_in[0];   // (512, 256)
    const float* w      = (const float*)d_in[1];   // (256, 256)
    const float* bias   = (const float*)d_in[2];   // (256,)
    const float* coeffs = (const float*)d_in[3];   // (256, 256, 15, 4)
    float* out          = (float*)d_out;           // (512, 256)

    dim3 grid(OUTF * (BATCH / BT));   // 512 blocks: (output feature) x (batch half)
    dim3 block(THREADS);              // 8 wave32 waves
    kan_layer_kernel<<<grid, block, 0, stream>>>(x, w, bias, coeffs, out);
}